// NetworkForAGraphWithAttributes_51608327029024
// MI455X (gfx1250) — compile-verified
//
#include <hip/hip_runtime.h>
#include <hip/hip_bf16.h>
#include <math.h>

#define N_NODES 10000
#define N_EDGES 160000
#define C_DIM   72
#define C_PAD   80      // padded N dimension (5 n-tiles of 16)
#define A_DIM   16
#define L_LAYERS 3
#define O_DIM   64
#define B_BATCH 16
#define SH_DIM  9
#define FC_DIM  100
#define WAVES_PER_BLOCK 8
#define SW_ELEMS (36 * C_PAD * 2)   // 36 k-pairs x 80 cols x 2 = 5760 floats
#define W_SRC    (C_DIM * C_DIM)    // 5184

typedef __attribute__((ext_vector_type(2))) float v2f;
typedef __attribute__((ext_vector_type(8))) float v8f;

__device__ __forceinline__ v8f wmma_f32x4(v2f a, v2f b, v8f c) {
  // D = A(16x4,f32) * B(4x16,f32) + C(16x16,f32)
  return __builtin_amdgcn_wmma_f32_16x16x4_f32(false, a, false, b, (short)0, c,
                                               false, false);
}

// ---------------------------------------------------------------------------
// h init: h[i][c] = (onehot @ W_in)[c] = W_in[0][c]
// ---------------------------------------------------------------------------
__global__ void init_h_kernel(const float* __restrict__ W_in,
                              float* __restrict__ h) {
  int gid = blockIdx.x * blockDim.x + threadIdx.x;
  if (gid >= N_NODES * C_DIM) return;
  int i = gid / C_DIM, c = gid % C_DIM;
  h[(size_t)i * C_PAD + c] = W_in[c];
}

// ---------------------------------------------------------------------------
// per-edge radial MLP * spherical harmonics -> path[e][s] for one layer
// ---------------------------------------------------------------------------
__global__ void edge_path_kernel(const float* __restrict__ pos,
                                 const int* __restrict__ esrc,
                                 const int* __restrict__ edst,
                                 const float* __restrict__ fc1_w,  // (L,1,FC)
                                 const float* __restrict__ fc1_b,  // (L,FC)
                                 const float* __restrict__ fc2_w,  // (L,FC,SH)
                                 int l, float* __restrict__ path) {
  int e = blockIdx.x * blockDim.x + threadIdx.x;
  if (e >= N_EDGES) return;
  int s = esrc[e], d = edst[e];
  float rx = pos[3 * s + 0] - pos[3 * d + 0];
  float ry = pos[3 * s + 1] - pos[3 * d + 1];
  float rz = pos[3 * s + 2] - pos[3 * d + 2];
  float r = sqrtf(rx * rx + ry * ry + rz * rz);
  float inv = 1.0f / fmaxf(r, 1e-9f);
  float x = rx * inv, y = ry * inv, z = rz * inv;
  const float s3 = 1.7320508075688772f, s5 = 2.23606797749979f,
              s15 = 3.872983346207417f;
  float sh[SH_DIM];
  sh[0] = 1.0f;
  sh[1] = s3 * x;
  sh[2] = s3 * y;
  sh[3] = s3 * z;
  sh[4] = s15 * x * y;
  sh[5] = s15 * y * z;
  sh[6] = 0.5f * s5 * (3.0f * z * z - 1.0f);
  sh[7] = s15 * x * z;
  sh[8] = 0.5f * s15 * (x * x - y * y);
  float t = r * 0.5f;  // r / MAX_RADIUS
  float emb = (t < 1.0f) ? cosf(1.5707963267948966f * t) : 0.0f;

  float u[SH_DIM];
#pragma unroll
  for (int j = 0; j < SH_DIM; ++j) u[j] = 0.0f;
  const float* w1 = fc1_w + l * FC_DIM;
  const float* b1 = fc1_b + l * FC_DIM;
  const float* w2 = fc2_w + (size_t)l * FC_DIM * SH_DIM;
  for (int f = 0; f < FC_DIM; ++f) {
    float a = emb * w1[f] + b1[f];
    float v = a / (1.0f + __expf(-a));  // silu
    const float* w2f = w2 + f * SH_DIM;
#pragma unroll
    for (int j = 0; j < SH_DIM; ++j) u[j] += v * w2f[j];
  }
  float* pe = path + (size_t)e * SH_DIM;
#pragma unroll
  for (int j = 0; j < SH_DIM; ++j) pe[j] = u[j] * sh[j];
}

// ---------------------------------------------------------------------------
// message kernel: per 16-edge tile, m = sum_s (diag(path_s) * H[src]) @ W_tp[l,s]
// WMMA f32 16x16x4; k-pair-interleaved LDS layout for B (one ds_load_b64 per
// fragment); double-buffered W staging through registers (one barrier per s).
// ---------------------------------------------------------------------------
union SMemU {
  float hs[WAVES_PER_BLOCK][16 * C_DIM];  // 36864 B, dead after A extraction
  float w[SW_ELEMS];                      // 23040 B, W buffer 1
};

__global__ __launch_bounds__(32 * WAVES_PER_BLOCK) void message_kernel(
    const float* __restrict__ h, const int* __restrict__ esrc,
    const int* __restrict__ edst, const float* __restrict__ path,
    const float* __restrict__ Wtp,  // (L,SH,C,C)
    int l, float* __restrict__ agg) {
  __shared__ float sW0[SW_ELEMS];  // 23040 B, W buffer 0
  __shared__ SMemU su;             // 36864 B
  __shared__ int sDst[WAVES_PER_BLOCK][16];

  const int tid = threadIdx.x;
  const int wave = tid >> 5;
  const int lane = tid & 31;
  const int row = lane & 15;  // A row / D column within 16-tile
  const int hi = lane >> 4;   // lane half
  const int klo = hi * 2;

  const int tile = blockIdx.x * WAVES_PER_BLOCK + wave;
  const int eb = tile * 16;

  // zero the pad columns (c=72..79) of buffer 0 once; they persist
  for (int idx = tid; idx < 36 * 8 * 2; idx += 32 * WAVES_PER_BLOCK) {
    int kp = idx / 16, r = idx % 16;
    sW0[(kp * C_PAD + 72 + (r >> 1)) * 2 + (r & 1)] = 0.0f;
  }

  // ---- stage this wave's 16 gathered h rows into LDS (b128 loads) ----
  for (int i = lane; i < 16 * (C_DIM / 4); i += 32) {
    int r = i / (C_DIM / 4);
    int c4 = (i % (C_DIM / 4)) * 4;
    const float4 v = *(const float4*)&h[(size_t)esrc[eb + r] * C_PAD + c4];
    *(float4*)&su.hs[wave][r * C_DIM + c4] = v;
  }
  if (lane < 16) sDst[wave][lane] = edst[eb + lane];

  // per-lane path values for this row (9 scalars)
  float pv[SH_DIM];
  {
    const float* pe = path + (size_t)(eb + row) * SH_DIM;
#pragma unroll
    for (int s = 0; s < SH_DIM; ++s) pv[s] = pe[s];
  }
  __syncthreads();

  // ---- pull A base fragments (unscaled) into registers: 18 k-steps x 2 ----
  float abase[36];
#pragma unroll
  for (int t = 0; t < 18; ++t) {
    const float2 av = *(const float2*)&su.hs[wave][row * C_DIM + 4 * t + klo];
    abase[2 * t] = av.x;
    abase[2 * t + 1] = av.y;
  }

  // stage W[s=0] into buffer 0 (no alias with hs region)
  {
    const float* W0 = Wtp + ((size_t)l * SH_DIM + 0) * W_SRC;
    for (int srci = tid; srci < W_SRC; srci += 32 * WAVES_PER_BLOCK) {
      int k = srci / C_DIM, c = srci % C_DIM;
      sW0[(k >> 1) * (C_PAD * 2) + c * 2 + (k & 1)] = W0[srci];
    }
  }
  __syncthreads();  // hs now dead everywhere; sW0 ready

  v8f acc[5];
#pragma unroll
  for (int nt = 0; nt < 5; ++nt) acc[nt] = v8f{};

  for (int s = 0; s < SH_DIM; ++s) {
    // issue next-W global loads early; latency hides under the WMMA block
    float wtmp[21];
    if (s < SH_DIM - 1) {
      const float* Wn = Wtp + ((size_t)l * SH_DIM + (s + 1)) * W_SRC;
#pragma unroll
      for (int i = 0; i < 21; ++i) {
        int srci = tid + i * (32 * WAVES_PER_BLOCK);
        if (srci < W_SRC) wtmp[i] = Wn[srci];
      }
    }
    if (s == 0) {
      // buffer 1 aliases hs; its pad columns can be zeroed only now
      for (int idx = tid; idx < 36 * 8 * 2; idx += 32 * WAVES_PER_BLOCK) {
        int kp = idx / 16, r = idx % 16;
        su.w[(kp * C_PAD + 72 + (r >> 1)) * 2 + (r & 1)] = 0.0f;
      }
    }

    const float* buf = (s & 1) ? su.w : sW0;
    const float p = pv[s];
#pragma unroll
    for (int nt = 0; nt < 5; ++nt) {
      const float* wb = buf + (nt * 16 + row) * 2;
#pragma unroll
      for (int t = 0; t < 18; ++t) {
        v2f a;
        a.x = abase[2 * t] * p;
        a.y = abase[2 * t + 1] * p;
        const float2 bv = *(const float2*)&wb[(2 * t + hi) * (C_PAD * 2)];
        v2f b;
        b.x = bv.x;
        b.y = bv.y;
        acc[nt] = wmma_f32x4(a, b, acc[nt]);
      }
    }

    // dump staged W into the other buffer
    if (s < SH_DIM - 1) {
      float* nbuf = (s & 1) ? sW0 : su.w;
#pragma unroll
      for (int i = 0; i < 21; ++i) {
        int srci = tid + i * (32 * WAVES_PER_BLOCK);
        if (srci < W_SRC) {
          int k = srci / C_DIM, c = srci % C_DIM;
          nbuf[(k >> 1) * (C_PAD * 2) + c * 2 + (k & 1)] = wtmp[i];
        }
      }
    }
    __syncthreads();  // all reads of buf done; next buffer fully written
  }

  // ---- scatter-add into agg (scaled by 1/sqrt(NUM_NEIGHBORS)) ----
  const float scale = 0.25f;
#pragma unroll
  for (int nt = 0; nt < 5; ++nt) {
    int c0 = nt * 16 + row;
    if (c0 < C_DIM) {
#pragma unroll
      for (int j = 0; j < 8; ++j) {
        int r = j + 8 * hi;  // D row
        int dn = sDst[wave][r];
        atomicAdd(&agg[(size_t)dn * C_PAD + c0], acc[nt][j] * scale);
      }
    }
  }
}

// ---------------------------------------------------------------------------
// node update: h_new = gelu(agg + h@W_self[l] + attr@W_attr[l])
// ---------------------------------------------------------------------------
__global__ void update_kernel(const float* __restrict__ hcur,
                              const float* __restrict__ agg,
                              const float* __restrict__ attr,
                              const float* __restrict__ W_self,  // (L,C,C)
                              const float* __restrict__ W_attr,  // (L,A,C)
                              int l, float* __restrict__ hnxt) {
  int gid = blockIdx.x * blockDim.x + threadIdx.x;
  if (gid >= N_NODES * C_DIM) return;
  int i = gid / C_DIM, c = gid % C_DIM;
  float x = agg[(size_t)i * C_PAD + c];
  const float* hr = hcur + (size_t)i * C_PAD;
  const float* ws = W_self + (size_t)l * C_DIM * C_DIM + c;
  for (int k = 0; k < C_DIM; ++k) x += hr[k] * ws[k * C_DIM];
  const float* ar = attr + (size_t)i * A_DIM;
  const float* wa = W_attr + (size_t)l * A_DIM * C_DIM + c;
#pragma unroll
  for (int a = 0; a < A_DIM; ++a) x += ar[a] * wa[a * C_DIM];
  // gelu (tanh approximation, jax default)
  float x3 = x * x * x;
  float g =
      0.5f * x * (1.0f + tanhf(0.7978845608028654f * (x + 0.044715f * x3)));
  hnxt[(size_t)i * C_PAD + c] = g;
}

// ---------------------------------------------------------------------------
// output projection + batch pooling: out[b] += (h@W_out)/sqrt(625)
// ---------------------------------------------------------------------------
__global__ void output_kernel(const float* __restrict__ h,
                              const float* __restrict__ W_out,  // (C,O)
                              const int* __restrict__ batch,
                              float* __restrict__ out) {
  int gid = blockIdx.x * blockDim.x + threadIdx.x;
  if (gid >= N_NODES * O_DIM) return;
  int i = gid / O_DIM, o = gid % O_DIM;
  const float* hr = h + (size_t)i * C_PAD;
  float acc = 0.0f;
  for (int c = 0; c < C_DIM; ++c) acc += hr[c] * W_out[c * O_DIM + o];
  atomicAdd(&out[batch[i] * O_DIM + o], acc * 0.04f);  // 1/sqrt(625)
}

// ---------------------------------------------------------------------------
extern "C" void kernel_launch(void* const* d_in, const int* in_sizes, int n_in,
                              void* d_out, int out_size, void* d_ws,
                              size_t ws_size, hipStream_t stream) {
  const float* pos = (const float*)d_in[0];
  const float* node_attr = (const float*)d_in[1];
  const int* batch = (const int*)d_in[2];
  const int* edge_src = (const int*)d_in[3];
  const int* edge_dst = (const int*)d_in[4];
  const float* W_in = (const float*)d_in[5];
  const float* W_tp = (const float*)d_in[6];
  const float* W_self = (const float*)d_in[7];
  const float* W_attr = (const float*)d_in[8];
  const float* fc1_w = (const float*)d_in[9];
  const float* fc1_b = (const float*)d_in[10];
  const float* fc2_w = (const float*)d_in[11];
  const float* W_out = (const float*)d_in[12];
  float* out = (float*)d_out;

  char* ws = (char*)d_ws;
  const size_t H_BYTES = (size_t)N_NODES * C_PAD * sizeof(float);  // 3.2 MB
  float* hA = (float*)(ws);
  float* hB = (float*)(ws + H_BYTES);
  float* agg = (float*)(ws + 2 * H_BYTES);
  float* path = (float*)(ws + 3 * H_BYTES);  // E*9*4 = 5.76 MB

  init_h_kernel<<<(N_NODES * C_DIM + 255) / 256, 256, 0, stream>>>(W_in, hA);

  float* cur = hA;
  float* nxt = hB;
  const int msg_blocks = (N_EDGES / 16) / WAVES_PER_BLOCK;  // 1250
  for (int l = 0; l < L_LAYERS; ++l) {
    edge_path_kernel<<<(N_EDGES + 255) / 256, 256, 0, stream>>>(
        pos, edge_src, edge_dst, fc1_w, fc1_b, fc2_w, l, path);
    hipMemsetAsync(agg, 0, H_BYTES, stream);
    message_kernel<<<msg_blocks, 32 * WAVES_PER_BLOCK, 0, stream>>>(
        cur, edge_src, edge_dst, path, W_tp, l, agg);
    update_kernel<<<(N_NODES * C_DIM + 255) / 256, 256, 0, stream>>>(
        cur, agg, node_attr, W_self, W_attr, l, nxt);
    float* t = cur;
    cur = nxt;
    nxt = t;
  }

  hipMemsetAsync(out, 0, (size_t)out_size * sizeof(float), stream);
  output_kernel<<<(N_NODES * O_DIM + 255) / 256, 256, 0, stream>>>(
      cur, W_out, batch, out);
  (void)in_sizes;
  (void)n_in;
  (void)ws_size;
}